// FeatureTokenizer_1357209666327
// MI455X (gfx1250) — compile-verified
//
#include <hip/hip_runtime.h>

typedef __attribute__((ext_vector_type(2))) float v2f;
typedef __attribute__((ext_vector_type(4))) float v4f;
typedef __attribute__((ext_vector_type(8))) float v8f;

#define NN    4096
#define CC    256
#define HIN   64
#define HOUT  128
#define ROWS  128           // rows of N per block
#define PAD   68            // LDS row stride in floats (64 + 4 pad -> conflict-free b64 reads)

__global__ __launch_bounds__(256)
void dwconv_wmma_f32_kernel(const float* __restrict__ x,
                            const float* __restrict__ w,
                            const float* __restrict__ bias,
                            float* __restrict__ out) {
    __shared__ float ldsX[ROWS * PAD];   // X tile: 128 rows x 64 (padded)
    __shared__ float ldsW[HOUT * PAD];   // W_c  : 128 rows x 64 (padded)

    const int c       = blockIdx.y;
    const int rowBase = blockIdx.x * ROWS;
    const int t       = threadIdx.x;

    // ---- stage X tile and W tile into LDS (coalesced float4) ----
    const float* xs = x + (size_t)rowBase * (CC * HIN) + (size_t)c * HIN;
    const float* ws = w + (size_t)c * (HOUT * HIN);
#pragma unroll
    for (int k = 0; k < 8; ++k) {
        int id   = t + k * 256;          // 0..2047 float4 slots
        int row  = id >> 4;              // 0..127
        int col4 = (id & 15) << 2;       // 0,4,...,60
        v4f vx = *(const v4f*)(xs + (size_t)row * (CC * HIN) + col4);
        *(v4f*)(&ldsX[row * PAD + col4]) = vx;
        v4f vw = *(const v4f*)(ws + row * HIN + col4);
        *(v4f*)(&ldsW[row * PAD + col4]) = vw;
    }
    __syncthreads();

    const int lane = t & 31;             // wave32
    const int wave = t >> 5;             // 8 waves per block
    const int half = lane >> 4;          // 0: lanes 0-15, 1: lanes 16-31
    const int lr   = lane & 15;

    // ---- accumulators initialized with bias (broadcast per output column) ----
    v8f acc[8];
#pragma unroll
    for (int j = 0; j < 8; ++j) {
        float bv = bias[c * HOUT + j * 16 + lr];
#pragma unroll
        for (int e = 0; e < 8; ++e) acc[j][e] = bv;
    }

    // A fragment source: lane holds X[rowBase + wave*16 + lr][4*kk + 2*half + {0,1}]
    const float* aBase = &ldsX[(wave * 16 + lr) * PAD + half * 2];
    // B fragment source: lane holds W[j*16 + lr][4*kk + 2*half + {0,1}]  (B[k][o] = w[o][k])
    const float* bBase = &ldsW[lr * PAD + half * 2];

#pragma unroll
    for (int kk = 0; kk < 16; ++kk) {            // K = 64 in steps of 4
        v2f a = *(const v2f*)(aBase + kk * 4);
#pragma unroll
        for (int j = 0; j < 8; ++j) {            // 8 column tiles of 16
            v2f bfrag = *(const v2f*)(bBase + j * 16 * PAD + kk * 4);
            acc[j] = __builtin_amdgcn_wmma_f32_16x16x4_f32(
                /*neg_a=*/false, a, /*neg_b=*/false, bfrag,
                /*c_mod=*/(short)0, acc[j], /*reuse_a=*/false, /*reuse_b=*/false);
        }
    }

    // ---- store: D element [M = v + 8*half][N = lr] of tile j -> out[n, c, j*16 + lr]
    float* outBase = out + (size_t)(rowBase + wave * 16 + half * 8) * (CC * HOUT)
                         + (size_t)c * HOUT + lr;
#pragma unroll
    for (int v = 0; v < 8; ++v) {
        float* orow = outBase + (size_t)v * (CC * HOUT);
#pragma unroll
        for (int j = 0; j < 8; ++j) {
            orow[j * 16] = acc[j][v];
        }
    }
}

extern "C" void kernel_launch(void* const* d_in, const int* in_sizes, int n_in,
                              void* d_out, int out_size, void* d_ws, size_t ws_size,
                              hipStream_t stream) {
    const float* x    = (const float*)d_in[0];   // [N, C, H_IN]  f32
    const float* w    = (const float*)d_in[1];   // [C, H_OUT, H_IN] f32
    const float* bias = (const float*)d_in[2];   // [C, H_OUT] f32
    float* out        = (float*)d_out;           // [N, C, H_OUT] f32

    dim3 grid(NN / ROWS, CC);   // 32 x 256 blocks
    dwconv_wmma_f32_kernel<<<grid, 256, 0, stream>>>(x, w, bias, out);
}